// ARMSNorm_19765439496654
// MI455X (gfx1250) — compile-verified
//
#include <hip/hip_runtime.h>

typedef float v2f __attribute__((ext_vector_type(2)));
typedef float v4f __attribute__((ext_vector_type(4)));
typedef float v8f __attribute__((ext_vector_type(8)));

#define DDIM 2048

// fma-refined division by a uniform scalar: q = x*r; q += (x - q*s)*r.
// Correctly rounded for the normal-range operands this kernel sees,
// 3 VALU ops instead of the ~10-op IEEE div expansion.
__device__ __forceinline__ float fdiv_u(float x, float s, float r) {
    float q = x * r;
    return fmaf(fmaf(-q, s, x), r, q);
}

// ---------------- k0: zero the atomic header ----------------
__global__ void armsn_init(unsigned* hdr) {
    hdr[0] = 0u;   // bits of max|x|
    hdr[1] = 0u;   // bits of max|y|
}

// ---------------- k1: global max |x| ----------------
__global__ void armsn_absmax(const v4f* __restrict__ x, int n4, unsigned* hdr) {
    float m = 0.0f;
    for (int i = blockIdx.x * blockDim.x + threadIdx.x; i < n4;
         i += gridDim.x * blockDim.x) {
        v4f v = x[i];
        m = fmaxf(m, fmaxf(fmaxf(fabsf(v.x), fabsf(v.y)),
                           fmaxf(fabsf(v.z), fabsf(v.w))));
    }
    #pragma unroll
    for (int off = 16; off > 0; off >>= 1)
        m = fmaxf(m, __shfl_xor(m, off, 32));
    __shared__ float sm[8];
    const int wave = threadIdx.x >> 5;
    const int lane = threadIdx.x & 31;
    if (lane == 0) sm[wave] = m;
    __syncthreads();
    if (threadIdx.x == 0) {
        float b = sm[0];
        #pragma unroll
        for (int w = 1; w < 8; ++w) b = fmaxf(b, sm[w]);
        atomicMax(&hdr[0], __float_as_uint(b));  // b >= 0: uint order == float order
    }
}

// ---------------- k2: per-row stats via WMMA row-sum ----------------
// Block = 256 threads (8 waves), handles 16 rows. Wave w owns columns
// [w*256, w*256+256). A-matrix (16x4 f32): lane<16 -> row=lane, K0/K1;
// lane>=16 -> row=lane-16, K2/K3. B = ones => D[m][*] = row-chunk sums
// (ones-B is lane-layout independent). Chunk sums <= 256*127^2 < 2^24,
// so f32 WMMA accumulation of the integer squares is exact.
__global__ void armsn_rowstats(const float* __restrict__ x,
                               const float* __restrict__ w,
                               float* __restrict__ inv_std_out,
                               unsigned* hdr) {
    __shared__ int   s_sum[128];   // [wave][row]
    __shared__ float s_max[128];
    const int tid  = threadIdx.x;
    const int wave = tid >> 5;
    const int lane = tid & 31;
    const int r    = lane & 15;
    const int koff = (lane >> 4) << 1;          // 0 or 2
    const int base_row = blockIdx.x * 16;

    const float scale_in = fmaxf(__uint_as_float(hdr[0]) * (1.0f / 127.0f), 1e-8f);
    const float rcp_in   = 1.0f / scale_in;

    const float* xrow = x + (size_t)(base_row + r) * DDIM;
    const int kbase = wave * 256;

    v8f c = {0.f, 0.f, 0.f, 0.f, 0.f, 0.f, 0.f, 0.f};
    const v2f ones = {1.0f, 1.0f};
    float wmax = 0.0f;

    #pragma unroll 4
    for (int t = 0; t < 64; ++t) {
        const int k = kbase + t * 4 + koff;
        v2f xv = *(const v2f*)(xrow + k);
        v2f wv = *(const v2f*)(w + k);
        float xi0 = fminf(fmaxf(rintf(fdiv_u(xv.x, scale_in, rcp_in)), -127.0f), 127.0f);
        float xi1 = fminf(fmaxf(rintf(fdiv_u(xv.y, scale_in, rcp_in)), -127.0f), 127.0f);
        v2f a;
        a.x = xi0 * xi0;                        // exact in f32 (<= 16129)
        a.y = xi1 * xi1;
        c = __builtin_amdgcn_wmma_f32_16x16x4_f32(
                false, a, false, ones, (short)0, c, false, false);
        wmax = fmaxf(wmax, fmaxf(fabsf(wv.x * xi0), fabsf(wv.y * xi1)));
    }

    // row max: lanes L and L+16 share a row
    wmax = fmaxf(wmax, __shfl_xor(wmax, 16, 32));
    if (lane < 16) s_max[wave * 16 + r] = wmax;
    // row sums: D vgpr j -> M=j (lanes 0-15) / M=j+8 (lanes 16-31); exact ints
    if (lane == 0) {
        #pragma unroll
        for (int j = 0; j < 8; ++j) s_sum[wave * 16 + j] = (int)c[j];
    } else if (lane == 16) {
        #pragma unroll
        for (int j = 0; j < 8; ++j) s_sum[wave * 16 + 8 + j] = (int)c[j];
    }
    __syncthreads();

    if (tid < 16) {
        int   sum = 0;
        float m   = 0.0f;
        #pragma unroll
        for (int wv = 0; wv < 8; ++wv) {
            sum += s_sum[wv * 16 + tid];
            m    = fmaxf(m, s_max[wv * 16 + tid]);
        }
        float mean = (float)sum * (scale_in * scale_in) * (1.0f / (float)DDIM);
        float varf = fminf(fmaxf(rintf(mean), 1.0f), 65535.0f);
        int d = (int)varf;
        int q = (int)sqrtf((float)d);           // floor sqrt (with fixups)
        if ((q + 1) * (q + 1) <= d) ++q;
        if (q * q > d) --q;
        int std_i = (d > q * q + q) ? (q + 1) : q;   // round-to-nearest int sqrt
        float inv_std = 1.0f / (float)std_i;         // std_i >= 1 > EPS
        inv_std_out[base_row + tid] = inv_std;
        float cand = m * scale_in * inv_std;         // row's max|y|
        atomicMax(&hdr[1], __float_as_uint(cand));
    }
}

// ---------------- k3: recompute y, output fake-quant, NT traffic ----------------
__global__ void armsn_output(const v4f* __restrict__ x,
                             const float* __restrict__ w,
                             const float* __restrict__ inv_std,
                             const unsigned* __restrict__ hdr,
                             v4f* __restrict__ out, int n4) {
    const float scale_in  = fmaxf(__uint_as_float(hdr[0]) * (1.0f / 127.0f), 1e-8f);
    const float scale_out = fmaxf(__uint_as_float(hdr[1]) * (1.0f / 127.0f), 1e-8f);
    const float rcp_in  = 1.0f / scale_in;
    const float rcp_out = 1.0f / scale_out;
    for (int i = blockIdx.x * blockDim.x + threadIdx.x; i < n4;
         i += gridDim.x * blockDim.x) {
        const int row  = i >> 9;                 // 512 float4 per 2048-row
        const int col  = (i & 511) << 2;
        const float is = inv_std[row];
        v4f v = __builtin_nontemporal_load(&x[i]);   // last use of x: let L2 drop it
        v4f wv = *(const v4f*)(w + col);
        v4f o;
        #pragma unroll
        for (int j = 0; j < 4; ++j) {
            float xi = fminf(fmaxf(rintf(fdiv_u(v[j], scale_in, rcp_in)), -127.0f), 127.0f);
            float y  = wv[j] * (xi * scale_in * is);
            o[j] = fminf(fmaxf(rintf(fdiv_u(y, scale_out, rcp_out)), -127.0f), 127.0f)
                   * scale_out;
        }
        __builtin_nontemporal_store(o, &out[i]); // streaming store, don't pollute L2
    }
}

extern "C" void kernel_launch(void* const* d_in, const int* in_sizes, int n_in,
                              void* d_out, int out_size, void* d_ws, size_t ws_size,
                              hipStream_t stream) {
    const float* x = (const float*)d_in[0];
    const float* w = (const float*)d_in[1];
    float* out = (float*)d_out;

    const int n    = in_sizes[0];        // 4*4096*2048
    const int rows = n / DDIM;           // 16384
    const int n4   = n / 4;

    unsigned* hdr     = (unsigned*)d_ws;
    float*    inv_std = (float*)((char*)d_ws + 256);   // rows*4 bytes of scratch

    armsn_init<<<1, 1, 0, stream>>>(hdr);
    armsn_absmax<<<4096, 256, 0, stream>>>((const v4f*)x, n4, hdr);
    armsn_rowstats<<<rows / 16, 256, 0, stream>>>(x, w, inv_std, hdr);
    armsn_output<<<4096, 256, 0, stream>>>((const v4f*)x, w, inv_std, hdr,
                                           (v4f*)out, n4);
}